// CVXPolicy_Integrator_14216341749946
// MI455X (gfx1250) — compile-verified
//
#include <hip/hip_runtime.h>
#include <hip/hip_bf16.h>

typedef float v2f __attribute__((ext_vector_type(2)));
typedef float v8f __attribute__((ext_vector_type(8)));

#define WAVES 8
#define ROWS_PER_BLOCK (WAVES * 16)

// LDS layouts:
//  W1p: W1 rows 1..256 (K=256 -> 128 K-pairs), N padded 100->112, pair-interleaved:
//       element (k,n), k = feature-1, at [((k>>1)*112 + n)*2 + (k&1)]
//  W2p: K=100 (50 pairs), N=256: element (k,n) at [((k>>1)*256 + n)*2 + (k&1)]
//  h1s: per-wave 16 x 104 fp32 scratch (tanh output), row stride 104 keeps 8B align
__global__ __launch_bounds__(256, 1)
void cvx_policy_kernel(const float* __restrict__ zin, const float* __restrict__ tin,
                       const float* __restrict__ W1, const float* __restrict__ b1,
                       const float* __restrict__ W2, const float* __restrict__ b2,
                       float* __restrict__ out)
{
    __shared__ float W1p[128 * 112 * 2];   // 114,688 B
    __shared__ float W2p[50 * 256 * 2];    // 102,400 B
    __shared__ float h1s[WAVES][16 * 104]; //  53,248 B   (total ~264 KB < 320 KB/WGP)

    const int tid = threadIdx.x;

    // ---- stage weights into LDS (pair-interleaved for b64 B-fragment loads) ----
    for (int idx = tid; idx < 128 * 112 * 2; idx += 256) W1p[idx] = 0.0f;
    __syncthreads();
    // W1 rows 1..256 only (row 0 = t-coefficients, folded into acc init)
    for (int idx = tid; idx < 256 * 100; idx += 256) {
        int k = idx / 100;                 // 0..255  (= feature-1)
        int n = idx - k * 100;
        W1p[((k >> 1) * 112 + n) * 2 + (k & 1)] = W1[(k + 1) * 100 + n];
    }
    for (int idx = tid; idx < 100 * 256; idx += 256) {
        int k = idx >> 8;
        int n = idx & 255;
        W2p[((k >> 1) * 256 + n) * 2 + (k & 1)] = W2[idx];
    }
    __syncthreads();

    const int lane = tid & 31;
    const int wave = tid >> 5;
    const int m    = lane & 15;   // A-layout: lane -> M row; B/C/D: lane&15 -> N col
    const int hi   = lane >> 4;   // half-wave selector (K-pair / M+8)
    const int rowBase = blockIdx.x * ROWS_PER_BLOCK + wave * 16;
    const long rowA   = rowBase + m;
    const float* zrow = zin + rowA * 256;

    // t values for the 8 rows this lane's C/D elements belong to (M = i + 8*hi)
    float tvals[8];
#pragma unroll
    for (int i = 0; i < 8; ++i) tvals[i] = tin[rowBase + 8 * hi + i];

    // ====== GEMM1: z[16 x 256] @ W1[1:,:] -> acc1 (7 N-tiles), init = t*W1[0]+b1
    v8f acc1[7];
#pragma unroll
    for (int nt = 0; nt < 7; ++nt) {
        int n  = nt * 16 + m;
        int nc = (n < 100) ? n : 99;       // clamp, then mask -> branchless
        float w1r0 = W1[nc];               // W1 row 0, column n
        float b1n  = b1[nc];
        if (n >= 100) { w1r0 = 0.0f; b1n = 0.0f; }
#pragma unroll
        for (int i = 0; i < 8; ++i)
            acc1[nt][i] = tvals[i] * w1r0 + b1n;
    }

#pragma unroll 1
    for (int ks = 0; ks < 64; ++ks) {
        const int kp = 2 * ks + hi;                    // K-pair index
        v2f aF = *(const v2f*)(zrow + 2 * kp);         // aligned b64, branch-free
        const float* wb = &W1p[(kp * 112 + m) * 2];
#pragma unroll
        for (int nt = 0; nt < 7; ++nt) {
            v2f bF = *(const v2f*)(wb + nt * 32);
            acc1[nt] = __builtin_amdgcn_wmma_f32_16x16x4_f32(
                false, aF, false, bF, (short)0, acc1[nt], false, false);
        }
    }

    // ---- tanh, reshape D-layout -> row-major LDS scratch ----
    float* h1w = h1s[wave];
#pragma unroll
    for (int nt = 0; nt < 7; ++nt) {
        int n = nt * 16 + m;
#pragma unroll
        for (int i = 0; i < 8; ++i) {
            float v = tanhf(acc1[nt][i]);  // padded cols: tanh(0) = 0
            if (n < 104) h1w[(i + 8 * hi) * 104 + n] = v;
        }
    }
    __syncthreads();

    // ====== GEMM2: h1[16 x 100] @ W2[100 x 256] -> acc2 (16 N-tiles) ==========
    v8f acc2[16];
    const v8f zero8 = {0.f, 0.f, 0.f, 0.f, 0.f, 0.f, 0.f, 0.f};
#pragma unroll
    for (int nt = 0; nt < 16; ++nt) acc2[nt] = zero8;

#pragma unroll 1
    for (int ks = 0; ks < 25; ++ks) {
        const int kp = 2 * ks + hi;
        v2f aF = *(const v2f*)&h1w[m * 104 + 2 * kp];
        const float* wb = &W2p[(kp * 256 + m) * 2];
#pragma unroll
        for (int nt = 0; nt < 16; ++nt) {
            v2f bF = *(const v2f*)(wb + nt * 32);
            acc2[nt] = __builtin_amdgcn_wmma_f32_16x16x4_f32(
                false, aF, false, bF, (short)0, acc2[nt], false, false);
        }
    }

    // ---- add b2; accumulate per-row ||p||^2 partials ----
    float r2v[8];
#pragma unroll
    for (int i = 0; i < 8; ++i) r2v[i] = 0.0f;
#pragma unroll
    for (int nt = 0; nt < 16; ++nt) {
        float bb = b2[nt * 16 + m];
#pragma unroll
        for (int i = 0; i < 8; ++i) {
            float p = acc2[nt][i] + bb;
            acc2[nt][i] = p;
            r2v[i] += p * p;
        }
    }
    // reduce across the 16 lanes of each half (rows 0..7 in lanes 0-15,
    // rows 8..15 in lanes 16-31; xor masks 1..8 stay within each half)
#pragma unroll
    for (int mask = 1; mask <= 8; mask <<= 1) {
#pragma unroll
        for (int i = 0; i < 8; ++i)
            r2v[i] += __shfl_xor(r2v[i], mask, 32);
    }

    // ---- Lambert-W Newton (30 iters) per row; fp32, matches reference ----
    float sca[8];
#pragma unroll
    for (int i = 0; i < 8; ++i) {
        const float r2 = r2v[i];
        float w = log1pf(r2);
#pragma unroll 1
        for (int it = 0; it < 30; ++it) {
            float ew = expf(w);
            w = w - (w * ew - r2) / ((1.0f + w) * ew);
        }
        w = fmaxf(w, 0.0f);
        float tn = sqrtf(w);
        float r  = sqrtf(r2);
        sca[i] = (r > 1e-12f) ? (tn / fmaxf(r, 1e-12f)) : 1.0f;
    }

    // ---- ustar = -scale * p ; D-layout scatter to row-major output ----
#pragma unroll
    for (int nt = 0; nt < 16; ++nt) {
        int n = nt * 16 + m;
#pragma unroll
        for (int i = 0; i < 8; ++i) {
            long r = (long)(rowBase + i + 8 * hi);
            out[r * 256 + n] = -sca[i] * acc2[nt][i];
        }
    }
}

extern "C" void kernel_launch(void* const* d_in, const int* in_sizes, int n_in,
                              void* d_out, int out_size, void* d_ws, size_t ws_size,
                              hipStream_t stream) {
    const float* z  = (const float*)d_in[0];
    const float* t  = (const float*)d_in[1];
    const float* W1 = (const float*)d_in[2];
    const float* b1 = (const float*)d_in[3];
    const float* W2 = (const float*)d_in[4];
    const float* b2 = (const float*)d_in[5];
    float* out = (float*)d_out;

    const int rows = in_sizes[0] / 256;            // B = 131072
    const int grid = rows / ROWS_PER_BLOCK;        // 1024 blocks, exact cover
    cvx_policy_kernel<<<grid, 256, 0, stream>>>(z, t, W1, b1, W2, b2, out);
}